// ProcessGNN_44856638439531
// MI455X (gfx1250) — compile-verified
//
#include <hip/hip_runtime.h>
#include <hip/hip_bf16.h>
#include <math.h>

typedef __attribute__((ext_vector_type(2))) float v2f;
typedef __attribute__((ext_vector_type(8))) float v8f;

#define NEG_SLOPE 0.2f

__device__ __forceinline__ float elu1(float v) { return v > 0.0f ? v : (expf(v) - 1.0f); }

__device__ __forceinline__ void atomicMaxF(float* addr, float v) {
  // IEEE sign-split trick; addr pre-initialized to -inf
  if (v >= 0.0f) atomicMax((int*)addr, __float_as_int(v));
  else           atomicMin((unsigned int*)addr, __float_as_uint(v));
}

__device__ __forceinline__ void edge_sd(int e, int E, const int* __restrict__ ei, int& s, int& d) {
  if (e < E) { s = ei[e]; d = ei[E + e]; } else { s = d = e - E; }
}

__device__ __forceinline__ void edge_info(int e, int E, const int* __restrict__ ei,
                                          const float* __restrict__ eattr,
                                          const float* __restrict__ lattr,
                                          int& s, int& d, float& a0, float& a1) {
  if (e < E) { s = ei[e]; d = ei[E + e]; a0 = eattr[2 * e]; a1 = eattr[2 * e + 1]; }
  else { s = d = e - E; a0 = lattr[2 * (e - E)]; a1 = lattr[2 * (e - E) + 1]; }
}

// ---------------- init scratch ----------------
__global__ void k_init(float* cnt, float* sums, float* amax1, float* den1, float* out1,
                       float* amax2, float* den2, float* out2, int N) {
  const float ninf = __int_as_float(0xFF800000);
  long long stride = (long long)gridDim.x * blockDim.x;
  long long total = (long long)N * 256;
  for (long long t = (long long)blockIdx.x * blockDim.x + threadIdx.x; t < total; t += stride) {
    out1[t] = 0.0f;
    if (t < (long long)N * 32) out2[t] = 0.0f;
    if (t < (long long)N * 4) { amax1[t] = ninf; den1[t] = 0.0f; }
    if (t < (long long)N * 2) sums[t] = 0.0f;
    if (t < N) { cnt[t] = 0.0f; amax2[t] = ninf; den2[t] = 0.0f; }
  }
}

// ---------------- edge-attention coefficients: ce1[4][2], ce2[2] ----------------
__global__ void k_ce(const float* __restrict__ We1, const float* __restrict__ ae1,
                     const float* __restrict__ We2, const float* __restrict__ ae2,
                     float* __restrict__ ce) {
  int t = threadIdx.x;
  if (t < 8) {
    int h = t >> 1, j = t & 1;
    float s = 0.0f;
    for (int c = 0; c < 64; ++c) s += We1[j * 256 + h * 64 + c] * ae1[h * 64 + c];
    ce[h * 2 + j] = s;
  } else if (t < 10) {
    int j = t - 8;
    float s = 0.0f;
    for (int c = 0; c < 32; ++c) s += We2[j * 32 + c] * ae2[c];
    ce[8 + j] = s;
  }
}

// ---------------- self-loop attr (mean incoming) ----------------
__global__ void k_deg(const int* __restrict__ ei, const float* __restrict__ eattr,
                      float* cnt, float* sums, int E) {
  int e = blockIdx.x * blockDim.x + threadIdx.x;
  if (e >= E) return;
  int d = ei[E + e];
  atomicAdd(&cnt[d], 1.0f);
  atomicAdd(&sums[2 * d + 0], eattr[2 * e + 0]);
  atomicAdd(&sums[2 * d + 1], eattr[2 * e + 1]);
}

__global__ void k_loopattr(float* lattr, const float* cnt, int N) {
  int n = blockIdx.x * blockDim.x + threadIdx.x;
  if (n >= N) return;
  float c = fmaxf(cnt[n], 1.0f);
  lattr[2 * n + 0] /= c;
  lattr[2 * n + 1] /= c;
}

// ---------------- layer1 GEMM: h1[N,256] = x[N,8] @ W1[8,256], f32 WMMA ----------------
__global__ void __launch_bounds__(256) k_gemm1(const float* __restrict__ x,
                                               const float* __restrict__ W1,
                                               float* __restrict__ h1, int N) {
  int wave = threadIdx.x >> 5;
  int lane = threadIdx.x & 31;
  int row0 = blockIdx.x * 16;
  int M = lane & 15;
  int kb = (lane >> 4) * 2;           // A: K offset per lane half; B: K row per lane half
  int r = row0 + M; if (r >= N) r = N - 1;   // clamp loads, keep EXEC all-ones
  v2f a0, a1;
  a0.x = x[r * 8 + kb];     a0.y = x[r * 8 + kb + 1];      // K = 0..3
  a1.x = x[r * 8 + 4 + kb]; a1.y = x[r * 8 + 4 + kb + 1];  // K = 4..7
  const bool full = (row0 + 16 <= N);                       // wave-uniform
  for (int t = 0; t < 2; ++t) {
    int col0 = (wave * 2 + t) * 16;
    v2f b0, b1;
    b0.x = W1[(kb)     * 256 + col0 + M];
    b0.y = W1[(kb + 1) * 256 + col0 + M];
    b1.x = W1[(4 + kb)     * 256 + col0 + M];
    b1.y = W1[(4 + kb + 1) * 256 + col0 + M];
    v8f c = {};
    c = __builtin_amdgcn_wmma_f32_16x16x4_f32(false, a0, false, b0, (short)0, c, false, false);
    c = __builtin_amdgcn_wmma_f32_16x16x4_f32(false, a1, false, b1, (short)0, c, false, false);
    int coln = col0 + M;
    int rowbase = row0 + (lane >> 4) * 8;
    if (full) {
#pragma unroll
      for (int rr = 0; rr < 8; ++rr) h1[(rowbase + rr) * 256 + coln] = c[rr];
    } else {
      for (int rr = 0; rr < 8; ++rr) {
        int orow = rowbase + rr;
        if (orow < N) h1[orow * 256 + coln] = c[rr];
      }
    }
  }
}

// ---------------- layer1 node attention terms ----------------
__global__ void k_alpha_node1(const float* __restrict__ h1, const float* __restrict__ as,
                              const float* __restrict__ ad, float* asrc, float* adst, int N) {
  int i = blockIdx.x * blockDim.x + threadIdx.x;
  if (i >= N * 4) return;
  int n = i >> 2, h = i & 3;
  float s = 0.0f, d = 0.0f;
  for (int c = 0; c < 64; c += 4) {
    const float4 v = *(const float4*)(h1 + n * 256 + h * 64 + c);
    const float4 vs = *(const float4*)(as + h * 64 + c);
    const float4 vd = *(const float4*)(ad + h * 64 + c);
    s += v.x * vs.x + v.y * vs.y + v.z * vs.z + v.w * vs.w;
    d += v.x * vd.x + v.y * vd.y + v.z * vd.z + v.w * vd.w;
  }
  asrc[i] = s; adst[i] = d;
}

// ---------------- layer1 edge alpha + segment max ----------------
__global__ void k_edge_alpha1(const int* __restrict__ ei, const float* __restrict__ eattr,
                              const float* __restrict__ lattr, const float* __restrict__ asrc,
                              const float* __restrict__ adst, const float* __restrict__ ce,
                              float* aE, float* amax, int E, int Etot) {
  int i = blockIdx.x * blockDim.x + threadIdx.x;
  if (i >= Etot * 4) return;
  int e = i >> 2, h = i & 3;
  int s, d; float a0, a1;
  edge_info(e, E, ei, eattr, lattr, s, d, a0, a1);
  float a = asrc[s * 4 + h] + adst[d * 4 + h] + a0 * ce[h * 2] + a1 * ce[h * 2 + 1];
  a = a >= 0.0f ? a : NEG_SLOPE * a;
  aE[i] = a;
  atomicMaxF(&amax[d * 4 + h], a);
}

// ---------------- layer1 softmax denom (stores exp in place) ----------------
__global__ void k_edge_den1(const int* __restrict__ ei, float* aE, const float* __restrict__ amax,
                            float* den, int E, int Etot) {
  int i = blockIdx.x * blockDim.x + threadIdx.x;
  if (i >= Etot * 4) return;
  int e = i >> 2, h = i & 3;
  int s, d; edge_sd(e, E, ei, s, d);
  float w = expf(aE[i] - amax[d * 4 + h]);
  aE[i] = w;
  atomicAdd(&den[d * 4 + h], w);
}

// ---------------- layer1 message aggregation: 4 edges/block, float4 gather ----------------
__global__ void __launch_bounds__(256) k_agg1(const int* __restrict__ ei,
                                              const float* __restrict__ h1,
                                              const float* __restrict__ aE,
                                              const float* __restrict__ den,
                                              float* out1, int E, int Etot) {
  int e = blockIdx.x * 4 + (threadIdx.x >> 6);
  if (e >= Etot) return;
  int c0 = (threadIdx.x & 63) * 4;          // channel base, 4 contiguous channels
  int h = c0 >> 6;
  int s, d; edge_sd(e, E, ei, s, d);
  float w = aE[e * 4 + h] / (den[d * 4 + h] + 1e-16f);
  const float4 hv = *(const float4*)(h1 + (size_t)s * 256 + c0);
  float* o = out1 + (size_t)d * 256 + c0;
  atomicAdd(o + 0, hv.x * w);
  atomicAdd(o + 1, hv.y * w);
  atomicAdd(o + 2, hv.z * w);
  atomicAdd(o + 3, hv.w * w);
}

// ---------------- bias + BN(eval) + ELU, in place, float4 ----------------
__global__ void k_post1(float* hb, const float* __restrict__ b1, const float* __restrict__ g1,
                        const float* __restrict__ be1, int N) {
  int i = blockIdx.x * blockDim.x + threadIdx.x;
  if (i >= N * 64) return;
  int c0 = (i & 63) * 4;
  float4 v = *(float4*)(hb + (size_t)(i >> 6) * 256 + c0);
  const float sc = rsqrtf(1.0f + 1e-5f);
  float r[4] = {v.x, v.y, v.z, v.w};
#pragma unroll
  for (int j = 0; j < 4; ++j) {
    float t = (r[j] + b1[c0 + j]) * (g1[c0 + j] * sc) + be1[c0 + j];
    r[j] = elu1(t);
  }
  *(float4*)(hb + (size_t)(i >> 6) * 256 + c0) = make_float4(r[0], r[1], r[2], r[3]);
}

// ---------------- layer2 GEMM: h2[N,32] = hb[N,256] @ W2[256,32], f32 WMMA, W2 in LDS ----------------
__global__ void __launch_bounds__(256) k_gemm2(const float* __restrict__ hb,
                                               const float* __restrict__ W2,
                                               float* __restrict__ h2, int N) {
  __shared__ float sW[256 * 32];
  for (int i = threadIdx.x; i < 256 * 32 / 4; i += 256)
    *(float4*)(sW + i * 4) = *(const float4*)(W2 + i * 4);
  __syncthreads();
  int wave = threadIdx.x >> 5;
  int lane = threadIdx.x & 31;
  int row0 = blockIdx.x * 128 + wave * 16;
  int M = lane & 15;
  int kb = (lane >> 4) * 2;
  int r = row0 + M; if (r >= N) r = N - 1;
  const bool full = (row0 + 16 <= N);        // wave-uniform
  for (int t = 0; t < 2; ++t) {
    int col0 = t * 16;
    v8f c = {};
    for (int k0 = 0; k0 < 256; k0 += 4) {
      v2f a, b;
      a.x = hb[(size_t)r * 256 + k0 + kb];
      a.y = hb[(size_t)r * 256 + k0 + kb + 1];
      b.x = sW[(k0 + kb)     * 32 + col0 + M];
      b.y = sW[(k0 + kb + 1) * 32 + col0 + M];
      c = __builtin_amdgcn_wmma_f32_16x16x4_f32(false, a, false, b, (short)0, c, false, false);
    }
    int coln = col0 + M;
    int rowbase = row0 + (lane >> 4) * 8;
    if (full) {
#pragma unroll
      for (int rr = 0; rr < 8; ++rr) h2[(rowbase + rr) * 32 + coln] = c[rr];
    } else {
      for (int rr = 0; rr < 8; ++rr) {
        int orow = rowbase + rr;
        if (orow < N) h2[orow * 32 + coln] = c[rr];
      }
    }
  }
}

// ---------------- layer2 node attention terms ----------------
__global__ void k_alpha_node2(const float* __restrict__ h2, const float* __restrict__ as,
                              const float* __restrict__ ad, float* asrc, float* adst, int N) {
  int n = blockIdx.x * blockDim.x + threadIdx.x;
  if (n >= N) return;
  float s = 0.0f, d = 0.0f;
  for (int c = 0; c < 32; c += 4) {
    const float4 v = *(const float4*)(h2 + (size_t)n * 32 + c);
    const float4 vs = *(const float4*)(as + c);
    const float4 vd = *(const float4*)(ad + c);
    s += v.x * vs.x + v.y * vs.y + v.z * vs.z + v.w * vs.w;
    d += v.x * vd.x + v.y * vd.y + v.z * vd.z + v.w * vd.w;
  }
  asrc[n] = s; adst[n] = d;
}

// ---------------- layer2 edge alpha + segment max ----------------
__global__ void k_edge_alpha2(const int* __restrict__ ei, const float* __restrict__ eattr,
                              const float* __restrict__ lattr, const float* __restrict__ asrc,
                              const float* __restrict__ adst, const float* __restrict__ ce,
                              float* aE, float* amax, int E, int Etot) {
  int e = blockIdx.x * blockDim.x + threadIdx.x;
  if (e >= Etot) return;
  int s, d; float a0, a1;
  edge_info(e, E, ei, eattr, lattr, s, d, a0, a1);
  float a = asrc[s] + adst[d] + a0 * ce[8] + a1 * ce[9];
  a = a >= 0.0f ? a : NEG_SLOPE * a;
  aE[e] = a;
  atomicMaxF(&amax[d], a);
}

__global__ void k_edge_den2(const int* __restrict__ ei, float* aE, const float* __restrict__ amax,
                            float* den, int E, int Etot) {
  int e = blockIdx.x * blockDim.x + threadIdx.x;
  if (e >= Etot) return;
  int s, d; edge_sd(e, E, ei, s, d);
  float w = expf(aE[e] - amax[d]);
  aE[e] = w;
  atomicAdd(&den[d], w);
}

// ---------------- layer2 aggregation: 8 threads/edge, float4 gather ----------------
__global__ void __launch_bounds__(256) k_agg2(const int* __restrict__ ei,
                                              const float* __restrict__ h2,
                                              const float* __restrict__ aE,
                                              const float* __restrict__ den,
                                              float* out2, int E, int Etot) {
  int i = blockIdx.x * blockDim.x + threadIdx.x;
  if (i >= Etot * 8) return;
  int e = i >> 3;
  int c0 = (i & 7) * 4;
  int s, d; edge_sd(e, E, ei, s, d);
  float w = aE[e] / (den[d] + 1e-16f);
  const float4 hv = *(const float4*)(h2 + (size_t)s * 32 + c0);
  float* o = out2 + (size_t)d * 32 + c0;
  atomicAdd(o + 0, hv.x * w);
  atomicAdd(o + 1, hv.y * w);
  atomicAdd(o + 2, hv.z * w);
  atomicAdd(o + 3, hv.w * w);
}

// ---------------- finalize: BN+ELU -> emb, MLP head -> pred ----------------
__global__ void k_final(const float* __restrict__ out2, const float* __restrict__ b2,
                        const float* __restrict__ g2, const float* __restrict__ be2,
                        const float* __restrict__ Wp1, const float* __restrict__ bp1,
                        const float* __restrict__ Wp2, const float* __restrict__ bp2,
                        float* __restrict__ outp, int N) {
  int n = blockIdx.x * blockDim.x + threadIdx.x;
  if (n >= N) return;
  float emb[32];
  float sc = rsqrtf(1.0f + 1e-5f);
  for (int c = 0; c < 32; ++c) {
    float v = (out2[(size_t)n * 32 + c] + b2[c]) * (g2[c] * sc) + be2[c];
    v = elu1(v);
    emb[c] = v;
    outp[(long long)N + (long long)n * 32 + c] = v;
  }
  float z = bp2[0];
  for (int j = 0; j < 16; ++j) {
    float p = bp1[j];
    for (int c = 0; c < 32; ++c) p += emb[c] * Wp1[c * 16 + j];
    p = p > 0.0f ? p : 0.0f;
    z += p * Wp2[j];
  }
  outp[n] = 1.0f / (1.0f + expf(-z));
}

extern "C" void kernel_launch(void* const* d_in, const int* in_sizes, int n_in,
                              void* d_out, int out_size, void* d_ws, size_t ws_size,
                              hipStream_t stream) {
  const float* x     = (const float*)d_in[0];
  const int*   ei    = (const int*)  d_in[1];
  const float* eattr = (const float*)d_in[2];
  const float* W1    = (const float*)d_in[3];
  const float* We1   = (const float*)d_in[4];
  const float* as1   = (const float*)d_in[5];
  const float* ad1   = (const float*)d_in[6];
  const float* ae1   = (const float*)d_in[7];
  const float* b1    = (const float*)d_in[8];
  const float* g1    = (const float*)d_in[9];
  const float* be1   = (const float*)d_in[10];
  const float* W2    = (const float*)d_in[11];
  const float* We2   = (const float*)d_in[12];
  const float* as2   = (const float*)d_in[13];
  const float* ad2   = (const float*)d_in[14];
  const float* ae2   = (const float*)d_in[15];
  const float* b2    = (const float*)d_in[16];
  const float* g2    = (const float*)d_in[17];
  const float* be2   = (const float*)d_in[18];
  const float* Wp1   = (const float*)d_in[19];
  const float* bp1   = (const float*)d_in[20];
  const float* Wp2   = (const float*)d_in[21];
  const float* bp2   = (const float*)d_in[22];

  const int N = in_sizes[0] / 8;
  const int E = in_sizes[1] / 2;
  const int Etot = E + N;

  float* p = (float*)d_ws;
  float* cnt   = p; p += (size_t)N;
  float* lattr = p; p += (size_t)2 * N;      // sums, then in-place loop_attr
  float* h1    = p; p += (size_t)256 * N;
  float* asrc1 = p; p += (size_t)4 * N;
  float* adst1 = p; p += (size_t)4 * N;
  float* aE1   = p; p += (size_t)4 * Etot;
  float* amax1 = p; p += (size_t)4 * N;
  float* den1  = p; p += (size_t)4 * N;
  float* hb    = p; p += (size_t)256 * N;    // out1 accumulator, then BN+ELU in place
  float* h2    = p; p += (size_t)32 * N;
  float* asrc2 = p; p += (size_t)N;
  float* adst2 = p; p += (size_t)N;
  float* aE2   = p; p += (size_t)Etot;
  float* amax2 = p; p += (size_t)N;
  float* den2  = p; p += (size_t)N;
  float* out2  = p; p += (size_t)32 * N;
  float* ce    = p; p += 16;

  const int B = 256;
  k_init<<<4096, B, 0, stream>>>(cnt, lattr, amax1, den1, hb, amax2, den2, out2, N);
  k_ce<<<1, 32, 0, stream>>>(We1, ae1, We2, ae2, ce);
  k_deg<<<(E + B - 1) / B, B, 0, stream>>>(ei, eattr, cnt, lattr, E);
  k_loopattr<<<(N + B - 1) / B, B, 0, stream>>>(lattr, cnt, N);

  // layer 1
  k_gemm1<<<(N + 15) / 16, B, 0, stream>>>(x, W1, h1, N);
  k_alpha_node1<<<((size_t)4 * N + B - 1) / B, B, 0, stream>>>(h1, as1, ad1, asrc1, adst1, N);
  k_edge_alpha1<<<((size_t)4 * Etot + B - 1) / B, B, 0, stream>>>(ei, eattr, lattr, asrc1, adst1, ce, aE1, amax1, E, Etot);
  k_edge_den1<<<((size_t)4 * Etot + B - 1) / B, B, 0, stream>>>(ei, aE1, amax1, den1, E, Etot);
  k_agg1<<<(Etot + 3) / 4, B, 0, stream>>>(ei, h1, aE1, den1, hb, E, Etot);
  k_post1<<<((size_t)64 * N + B - 1) / B, B, 0, stream>>>(hb, b1, g1, be1, N);

  // layer 2
  k_gemm2<<<(N + 127) / 128, B, 0, stream>>>(hb, W2, h2, N);
  k_alpha_node2<<<(N + B - 1) / B, B, 0, stream>>>(h2, as2, ad2, asrc2, adst2, N);
  k_edge_alpha2<<<(Etot + B - 1) / B, B, 0, stream>>>(ei, eattr, lattr, asrc2, adst2, ce, aE2, amax2, E, Etot);
  k_edge_den2<<<(Etot + B - 1) / B, B, 0, stream>>>(ei, aE2, amax2, den2, E, Etot);
  k_agg2<<<((size_t)8 * Etot + B - 1) / B, B, 0, stream>>>(ei, h2, aE2, den2, out2, E, Etot);

  // head
  k_final<<<(N + B - 1) / B, B, 0, stream>>>(out2, b2, g2, be2, Wp1, bp1, Wp2, bp2, (float*)d_out, N);
}